// NeuralRasterizationLayer_16484084482653
// MI455X (gfx1250) — compile-verified
//
#include <hip/hip_runtime.h>
#include <cstdint>

#define SIZE_PX 32
#define NPTS    256
#define NSEG    255
#define WIDTHF  0.5f
#define EPSF    1e-6f
#define TINYF   1e-12f

typedef __attribute__((ext_vector_type(2))) float v2f;
typedef __attribute__((ext_vector_type(8))) float v8f;

#if __has_builtin(__builtin_amdgcn_wmma_f32_16x16x4_f32)
#define HAVE_WMMA_F32 1
#else
#define HAVE_WMMA_F32 0
#endif

// One block = (batch b, pixel-quadrant q). 256 threads = 8 wave32.
// Phase 0: async-stage the batch's points/atts into LDS (CDNA5 async global->LDS).
// Phase 1: prefix-sum of point deltas as a triangular matmul on the WMMA pipe:
//          within-chunk inclusive scan Y = L16 x X via 4 chained V_WMMA_F32_16X16X4_F32,
//          then chunk-offset fixup. Waves 0/1 handle the two coordinate components.
// Phase 2: each thread owns one pixel; loop 255 segments with LDS-broadcast
//          segment state and wave-uniform early-outs.
__global__ __launch_bounds__(256) void NeuralRaster_kernel(
    const float* __restrict__ points, const float* __restrict__ atts,
    float* __restrict__ out)
{
    __shared__ float2 pts_l[NPTS];   // (points[...,0], points[...,1])
    __shared__ float2 att_l[NPTS];   // (pen, intensity)
    __shared__ float  sc0[NPTS];     // within-chunk scan of points[...,1]  -> px/SIZE
    __shared__ float  sc1[NPTS];     // within-chunk scan of points[...,0]  -> py/SIZE
    __shared__ float  pxl[NPTS];
    __shared__ float  pyl[NPTS];

    const int t = threadIdx.x;
    const int b = blockIdx.x;
    const int quad = blockIdx.y;

    // ---------------- Phase 0: async global -> LDS staging -------------------
    {
        uint64_t gp = (uint64_t)(uintptr_t)(points + ((size_t)b * NPTS + t) * 2);
        uint64_t ga = (uint64_t)(uintptr_t)(atts   + ((size_t)b * NPTS + t) * 2);
        uint32_t lp = (uint32_t)(uintptr_t)(&pts_l[t]);
        uint32_t la = (uint32_t)(uintptr_t)(&att_l[t]);
        asm volatile("global_load_async_to_lds_b64 %0, %1, off"
                     :: "v"(lp), "v"(gp) : "memory");
        asm volatile("global_load_async_to_lds_b64 %0, %1, off"
                     :: "v"(la), "v"(ga) : "memory");
        asm volatile("s_wait_asynccnt 0" ::: "memory");
    }
    __syncthreads();

    // ---------------- Phase 1: prefix sum (cumsum) ---------------------------
#if HAVE_WMMA_F32
    {
        const int wave = t >> 5;
        const int lane = t & 31;
        if (wave < 2) {
            const int c  = lane & 15;   // chunk id == B/D column, and A row (m)
            const int hi = lane >> 4;   // K half selector per ISA layout
            v8f acc = {0.f, 0.f, 0.f, 0.f, 0.f, 0.f, 0.f, 0.f};
#pragma unroll
            for (int kb = 0; kb < 4; ++kb) {
                const int kr = 4 * kb + 2 * hi;   // global K column of this VGPR pair
                const int n0 = c * 16 + kr;       // element index inside chunk c
                v2f afrag, bfrag;
                // A = L16[:, kr:kr+2] : lower-triangular ones (k <= m), m = c
                afrag.x = (kr     <= c) ? 1.f : 0.f;
                afrag.y = (kr + 1 <= c) ? 1.f : 0.f;
                // B = X[kr:kr+2, c] : X[k][c] = delta[c*16 + k]
                if (wave == 0) { bfrag.x = pts_l[n0].y; bfrag.y = pts_l[n0 + 1].y; }
                else           { bfrag.x = pts_l[n0].x; bfrag.y = pts_l[n0 + 1].x; }
                acc = __builtin_amdgcn_wmma_f32_16x16x4_f32(
                        false, afrag, false, bfrag, (short)0, acc, false, false);
            }
            // D[m][c], m = v + 8*hi  ->  scan value at n = c*16 + m
            float* dst = (wave == 0) ? sc0 : sc1;
#pragma unroll
            for (int v = 0; v < 8; ++v) dst[c * 16 + 8 * hi + v] = acc[v];
        }
    }
    __syncthreads();
    {
        // add exclusive chunk offsets, scale by SIZE
        const int chunk = t >> 4;
        float o0 = 0.f, o1 = 0.f;
        for (int cc = 0; cc < chunk; ++cc) {
            o0 += sc0[cc * 16 + 15];
            o1 += sc1[cc * 16 + 15];
        }
        pxl[t] = (sc0[t] + o0) * (float)SIZE_PX;
        pyl[t] = (sc1[t] + o1) * (float)SIZE_PX;
    }
    __syncthreads();
#else
    // Fallback: Hillis-Steele scan in LDS
    sc0[t] = pts_l[t].y;
    sc1[t] = pts_l[t].x;
    __syncthreads();
    for (int d = 1; d < NPTS; d <<= 1) {
        float a0 = (t >= d) ? sc0[t - d] : 0.f;
        float a1 = (t >= d) ? sc1[t - d] : 0.f;
        __syncthreads();
        sc0[t] += a0;
        sc1[t] += a1;
        __syncthreads();
    }
    pxl[t] = sc0[t] * (float)SIZE_PX;
    pyl[t] = sc1[t] * (float)SIZE_PX;
    __syncthreads();
#endif

    // ---------------- Phase 2: rasterize one pixel per thread ----------------
    const int   p   = quad * 256 + t;       // pixel id 0..1023
    const float ixf = (float)(p >> 5);
    const float iyf = (float)(p & 31);
    float vmax = 0.f;

    for (int s = 0; s < NSEG; ++s) {
        const float p0x = pxl[s],     p0y = pyl[s];
        const float p1x = pxl[s + 1], p1y = pyl[s + 1];
        const float pen1 = att_l[s + 1].x;
        const bool  cm = ((p1x != 0.f) && (p1y != 0.f)) ||
                         ((p0x != 0.f) && (p0y != 0.f));
        // wave-uniform early out (segment state is LDS-broadcast)
        if ((pen1 != 0.f) || !cm) continue;

        const bool cx = ((ixf <= p1x + WIDTHF) && (ixf >= p0x - WIDTHF)) ||
                        ((ixf >= p1x - WIDTHF) && (ixf <= p0x + WIDTHF));
        const bool cy = ((iyf <= p1y + WIDTHF) && (iyf >= p0y - WIDTHF)) ||
                        ((iyf >= p1y - WIDTHF) && (iyf <= p0y + WIDTHF));
        if (!(cx && cy)) continue;

        const float a0 = att_l[s].y, a1 = att_l[s + 1].y;
        const float d1 = p1x - p0x, d2 = p1y - p0y;
        const float denom = sqrtf(d1 * d1 + d2 * d2 + TINYF);
        const float inv   = 1.f / (denom + EPSF);
        const float cte   = p1y * p0x - p1x * p0y;
        const float f0 = (d1 == 0.f) ? 1.f : 0.f;
        const float f1 = (d2 == 0.f) ? 1.f : 0.f;

        const float en   = fabsf(d1 * iyf - d2 * ixf + cte);
        const float val  = en * inv;
        const float val0 = fabsf(ixf - p0x);
        const float val1 = fabsf(iyf - p0y);
        const float dist = f0 * val0 + f1 * val1 + (1.f - f0 - f1) * val;
        if (!(dist < WIDTHF)) continue;

        const float dx0 = ixf - p0x, dy0 = iyf - p0y;
        const float dx1 = ixf - p1x, dy1 = iyf - p1y;
        const float sp0 = sqrtf(dx0 * dx0 + dy0 * dy0 + TINYF);
        const float sp1 = sqrtf(dx1 * dx1 + dy1 * dy1 + TINYF);
        const float dsq = dist * dist;
        const float z0  = sp0 * sp0 - dsq;
        const float z1  = sp1 * sp1 - dsq;
        const float l0  = (z0 > 0.f) ? sqrtf(fmaxf(z0, TINYF)) : 0.f;
        const float l1  = (z1 > 0.f) ? sqrtf(fmaxf(z1, TINYF)) : 0.f;
        const float vv  = (l0 * a0 + l1 * a1) / (l0 + l1 + EPSF);
        vmax = fmaxf(vmax, vv);
    }

    out[(size_t)b * 1024 + p] = fminf(1.f, vmax) * 2.f - 1.f;
}

extern "C" void kernel_launch(void* const* d_in, const int* in_sizes, int n_in,
                              void* d_out, int out_size, void* d_ws, size_t ws_size,
                              hipStream_t stream) {
    (void)n_in; (void)out_size; (void)d_ws; (void)ws_size;
    const float* points = (const float*)d_in[0];   // (B, 256, 2) f32
    const float* atts   = (const float*)d_in[1];   // (B, 256, 2) f32
    float* out = (float*)d_out;                    // (B, 32, 32, 1) f32

    const int B = in_sizes[0] / (NPTS * 2);        // 64
    dim3 grid(B, 4);                               // 4 pixel quadrants per batch
    NeuralRaster_kernel<<<grid, 256, 0, stream>>>(points, atts, out);
}